// LearnerModel_15728170238455
// MI455X (gfx1250) — compile-verified
//
#include <hip/hip_runtime.h>
#include <hip/hip_bf16.h>
#include <math.h>

#define H 4096
#define V 50257

typedef __attribute__((ext_vector_type(2))) float v2f;
typedef __attribute__((ext_vector_type(8))) float v8f;

// ---------------------------------------------------------------------------
// WMMA-based GEMV accumulator.
// Each wave computes dot(W[row], x) for 16 consecutive rows [rowBase, rowBase+16)
// using V_WMMA_F32_16X16X4_F32, with the x-vector broadcast across all 16 N
// columns of the B fragment (so D[m][n] == dot(row m, x) for every n).
//
// A (16x4 f32) lane layout per ISA 7.12.2: lanes 0-15 hold M=lane, K={k,k+1};
// lanes 16-31 hold M=lane-16, K={k+2,k+3}.  B (4x16) mirrors that K pairing.
// Weight loads are nontemporal (streamed exactly once); vector loads stay
// cached (rebroadcast to every wave).
// ---------------------------------------------------------------------------
__device__ __forceinline__ v8f wmma_gemv_acc(const float* __restrict__ W,
                                             size_t ld,
                                             const float* __restrict__ x,
                                             int K, int rowBase, int rowMax,
                                             v8f c) {
  const int lane = threadIdx.x & 31;
  int row = rowBase + (lane & 15);
  if (row >= rowMax) row = rowMax - 1;          // clamp loads; EXEC stays all-1s
  const int koff = (lane >> 4) << 1;            // 0 for lanes 0-15, 2 for 16-31
  const float* wr = W + (size_t)row * ld + koff;
  const float* xr = x + koff;
#pragma unroll 8
  for (int k = 0; k < K; k += 4) {
    v2f a = __builtin_nontemporal_load((const v2f*)(wr + k));
    v2f b = *(const v2f*)(xr + k);
    // (neg_a, A, neg_b, B, c_mod, C, reuse_a, reuse_b)
    c = __builtin_amdgcn_wmma_f32_16x16x4_f32(false, a, false, b,
                                              (short)0, c, false, false);
  }
  return c;
}

// D layout (32-bit C/D 16x16): VGPR j -> M=j on lanes 0-15 (N=lane),
// M=8+j on lanes 16-31.  Lane 0 holds rows rowBase+0..7 in c[0..7],
// lane 16 holds rows rowBase+8..15.

// --- kernel 1: h1 = tanh(W_ih @ emb[token] + b_ih + b_hh) -------------------
__global__ void k_rnn1(const int* __restrict__ tok,
                       const float* __restrict__ emb,
                       const float* __restrict__ W_ih,
                       const float* __restrict__ b_ih,
                       const float* __restrict__ b_hh,
                       float* __restrict__ h1) {
  const int wave = (int)((blockIdx.x * blockDim.x + threadIdx.x) >> 5);
  const int rowBase = wave * 16;
  const float* x = emb + (size_t)tok[0] * H;
  v8f c = {};
  c = wmma_gemv_acc(W_ih, H, x, H, rowBase, H, c);
  const int lane = threadIdx.x & 31;
  if ((lane & 15) == 0) {
    const int mb = rowBase + ((lane >> 4) << 3);
#pragma unroll
    for (int j = 0; j < 8; ++j) {
      const int r = mb + j;
      h1[r] = tanhf(c[j] + b_ih[r] + b_hh[r]);
    }
  }
}

// --- kernel 2: x1 = relu(comb_W @ [h1; h1] + comb_b) ------------------------
// attn_weights == softmax(single logit) == 1, so attn_applied == h1 exactly.
__global__ void k_comb(const float* __restrict__ comb_W,
                       const float* __restrict__ comb_b,
                       const float* __restrict__ h1,
                       float* __restrict__ x1) {
  const int wave = (int)((blockIdx.x * blockDim.x + threadIdx.x) >> 5);
  const int rowBase = wave * 16;
  v8f c = {};
  c = wmma_gemv_acc(comb_W,      (size_t)(2 * H), h1, H, rowBase, H, c);
  c = wmma_gemv_acc(comb_W + H,  (size_t)(2 * H), h1, H, rowBase, H, c);
  const int lane = threadIdx.x & 31;
  if ((lane & 15) == 0) {
    const int mb = rowBase + ((lane >> 4) << 3);
#pragma unroll
    for (int j = 0; j < 8; ++j) {
      const int r = mb + j;
      x1[r] = fmaxf(0.0f, c[j] + comb_b[r]);
    }
  }
}

// --- kernel 3: h2 = tanh(W_ih @ x1 + W_hh @ h1 + b_ih + b_hh) ---------------
__global__ void k_rnn2(const float* __restrict__ W_ih,
                       const float* __restrict__ W_hh,
                       const float* __restrict__ b_ih,
                       const float* __restrict__ b_hh,
                       const float* __restrict__ x1,
                       const float* __restrict__ h1,
                       float* __restrict__ h2) {
  const int wave = (int)((blockIdx.x * blockDim.x + threadIdx.x) >> 5);
  const int rowBase = wave * 16;
  v8f c = {};
  c = wmma_gemv_acc(W_ih, H, x1, H, rowBase, H, c);
  c = wmma_gemv_acc(W_hh, H, h1, H, rowBase, H, c);
  const int lane = threadIdx.x & 31;
  if ((lane & 15) == 0) {
    const int mb = rowBase + ((lane >> 4) << 3);
#pragma unroll
    for (int j = 0; j < 8; ++j) {
      const int r = mb + j;
      h2[r] = tanhf(c[j] + b_ih[r] + b_hh[r]);
    }
  }
}

// --- kernel 4: logits = out_W @ h2 + out_b  (dominant: ~823 MB streamed) ----
__global__ void k_logits(const float* __restrict__ out_W,
                         const float* __restrict__ out_b,
                         const float* __restrict__ h2,
                         float* __restrict__ logits) {
  const int wave = (int)((blockIdx.x * blockDim.x + threadIdx.x) >> 5);
  const int rowBase = wave * 16;
  v8f c = {};
  c = wmma_gemv_acc(out_W, H, h2, H, rowBase, V, c);
  const int lane = threadIdx.x & 31;
  if ((lane & 15) == 0) {
    const int mb = rowBase + ((lane >> 4) << 3);
#pragma unroll
    for (int j = 0; j < 8; ++j) {
      const int r = mb + j;
      if (r < V) logits[r] = c[j] + out_b[r];
    }
  }
}

// --- kernel 5: lse = max(logits) + log(sum(exp(logits - max))) --------------
__global__ void k_lse(const float* __restrict__ logits, float* __restrict__ lse) {
  __shared__ float red[1024];
  const int t = threadIdx.x;
  float m = -INFINITY;
  for (int i = t; i < V; i += 1024) m = fmaxf(m, logits[i]);
  red[t] = m;
  __syncthreads();
  for (int s = 512; s > 0; s >>= 1) {
    if (t < s) red[t] = fmaxf(red[t], red[t + s]);
    __syncthreads();
  }
  m = red[0];
  __syncthreads();
  float ssum = 0.0f;
  for (int i = t; i < V; i += 1024) ssum += expf(logits[i] - m);
  red[t] = ssum;
  __syncthreads();
  for (int s = 512; s > 0; s >>= 1) {
    if (t < s) red[t] += red[t + s];
    __syncthreads();
  }
  if (t == 0) lse[0] = m + logf(red[0]);
}

// --- kernel 6: out = logits - lse -------------------------------------------
__global__ void k_final(const float* __restrict__ logits,
                        const float* __restrict__ lse,
                        float* __restrict__ out) {
  const int i = (int)(blockIdx.x * blockDim.x + threadIdx.x);
  if (i < V) out[i] = logits[i] - lse[0];
}

extern "C" void kernel_launch(void* const* d_in, const int* in_sizes, int n_in,
                              void* d_out, int out_size, void* d_ws, size_t ws_size,
                              hipStream_t stream) {
  const int*   tok    = (const int*)  d_in[0];
  const float* emb    = (const float*)d_in[1];
  const float* W_ih   = (const float*)d_in[2];
  const float* W_hh   = (const float*)d_in[3];
  const float* b_ih   = (const float*)d_in[4];
  const float* b_hh   = (const float*)d_in[5];
  // d_in[6]=attn_W, d_in[7]=attn_b: provably dead (softmax over 1 logit == 1)
  const float* comb_W = (const float*)d_in[8];
  const float* comb_b = (const float*)d_in[9];
  const float* out_W  = (const float*)d_in[10];
  const float* out_b  = (const float*)d_in[11];
  float* out = (float*)d_out;

  // workspace layout (floats): h1[H] | x1[H] | h2[H] | logits[V] | lse[1]
  float* ws     = (float*)d_ws;
  float* h1     = ws;
  float* x1     = ws + H;
  float* h2     = ws + 2 * H;
  float* logits = ws + 3 * H;
  float* lse    = ws + 3 * H + V;

  const int BLK = 256;                       // 8 wave32s per block
  const int ROWS_PER_BLK = (BLK / 32) * 16;  // 128 rows per block

  const int blocksH = (H + ROWS_PER_BLK - 1) / ROWS_PER_BLK;   // 32
  const int blocksV = (V + ROWS_PER_BLK - 1) / ROWS_PER_BLK;   // 393

  k_rnn1  <<<blocksH, BLK, 0, stream>>>(tok, emb, W_ih, b_ih, b_hh, h1);
  k_comb  <<<blocksH, BLK, 0, stream>>>(comb_W, comb_b, h1, x1);
  k_rnn2  <<<blocksH, BLK, 0, stream>>>(W_ih, W_hh, b_ih, b_hh, x1, h1, h2);
  k_logits<<<blocksV, BLK, 0, stream>>>(out_W, out_b, h2, logits);
  k_lse   <<<1, 1024, 0, stream>>>(logits, lse);
  k_final <<<(V + BLK - 1) / BLK, BLK, 0, stream>>>(logits, lse, out);
}